// HetroGIN_39582418600210
// MI455X (gfx1250) — compile-verified
//
#include <hip/hip_runtime.h>
#include <hip/hip_bf16.h>

// ---------------------------------------------------------------------------
// HeteroGIN forward for MI455X (gfx1250, wave32).
// Dense GEMMs use V_WMMA_F32_16X16X4_F32 (exact fp32, matrix pipe).
// Edge aggregation uses global f32 atomics (bandwidth-bound anyway).
// ---------------------------------------------------------------------------

typedef __attribute__((ext_vector_type(2))) float v2f;
typedef __attribute__((ext_vector_type(8))) float v8f;

#define P_CNT 200000
#define L_CNT 100000
#define N_CNT 20000
#define EPL   1000000
#define ELN   200000
#define D_EMB 64

// ---------------------------------------------------------------------------
// Scatter-add (generic F): agg[dst][f] += src[srcid][f] over E edges.
// ei layout: [2][E] row-major; ei[e] = source id, ei[E+e] = dest id.
// ---------------------------------------------------------------------------
__global__ void scatter_add_kernel(const float* __restrict__ src, int F,
                                   const int* __restrict__ ei, int E,
                                   float* __restrict__ agg) {
    long long t = (long long)blockIdx.x * blockDim.x + threadIdx.x;
    long long total = (long long)E * F;
    if (t >= total) return;
    int e = (int)(t / F);
    int f = (int)(t - (long long)e * F);
    int s = ei[e];
    int d = ei[E + e];
    atomicAdd(&agg[(long long)d * F + f], src[(long long)s * F + f]);
}

// Specialized F==64 scatter: 16 threads/edge, float4 payload per thread.
// 4x fewer edge-index loads and 4x fewer vmem load instructions.
__global__ void scatter_add64_kernel(const float* __restrict__ src,
                                     const int* __restrict__ ei, int E,
                                     float* __restrict__ agg) {
    long long t = (long long)blockIdx.x * blockDim.x + threadIdx.x;
    if (t >= (long long)E * 16) return;
    int e = (int)(t >> 4);
    int q = ((int)t & 15) << 2;
    int s = ei[e];
    int d = ei[E + e];
    float4 v = *(const float4*)(src + (size_t)s * 64 + q);
    float* dst = agg + (size_t)d * 64 + q;
    atomicAdd(dst + 0, v.x);
    atomicAdd(dst + 1, v.y);
    atomicAdd(dst + 2, v.z);
    atomicAdd(dst + 3, v.w);
}

// ---------------------------------------------------------------------------
// Generic fused GIN/MLP GEMM, fp32 WMMA.
//   h (concat mode, sum_mode==0): h[k] = k<K1 ? A1[k] : ope*A2[k-K1], K = K1+K2
//   h (sum    mode, sum_mode==1): h[k] = A1[k] + ope*A2[k],           K = K1
//   out[row, col] (+)= act(h @ W + bias), act = PReLU(slope) (slope=0 -> ReLU)
// Block: 128 threads (4 waves). Tile: 64 rows x 64 cols. K chunked by 64.
// As: [64 rows][68] floats  -> a-frag ds_load b64, conflict-free.
// Wp: [32 k-pairs][80] f2   -> b-frag is ONE ds_load_b64 per tile; row stride
//     80*2=160 floats == 32 mod 64 banks, so the two half-waves (pair rows r
//     and r+1) hit disjoint bank halves: fully conflict-free.
// Wave w owns rows [16w,16w+16); A-frag reused across 4 column tiles.
// ---------------------------------------------------------------------------
__global__ __launch_bounds__(128)
void gemm_wmma_kernel(const float* __restrict__ A1, int ldA1, int K1,
                      const float* __restrict__ A2, int ldA2, int K2,
                      const float* __restrict__ eps_p,   // nullable, ope = 1+*eps_p
                      int sum_mode,
                      const float* __restrict__ W, int Nfull,
                      const float* __restrict__ bias,
                      const float* __restrict__ slope_p, float slope_c,
                      float* __restrict__ out, int ldo,
                      int rows, int accumulate) {
    __shared__ float  As[64 * 68];   // 17408 B
    __shared__ float2 Wp[32 * 80];   // 20480 B

    const int tid    = threadIdx.x;
    const int wave   = tid >> 5;
    const int lane   = tid & 31;
    const int lane16 = lane & 15;
    const int hi     = lane >> 4;

    const int rowbase = blockIdx.x * 64;
    const int colbase = blockIdx.y * 64;

    float ope   = 1.0f + (eps_p ? *eps_p : 0.0f);
    float slope = slope_p ? *slope_p : slope_c;

    const int K = sum_mode ? K1 : (K1 + K2);

    v8f acc[4] = {};

    for (int k0 = 0; k0 < K; k0 += 64) {
        const int kc  = min(64, K - k0);
        const int kcp = (kc + 3) & ~3;

        // --- stage h tile (64 rows x 64 k) ---
        for (int idx = tid; idx < 64 * 64; idx += 128) {
            int r = idx >> 6, k = idx & 63;
            float v = 0.0f;
            int row = rowbase + r;
            int kg  = k0 + k;
            if (row < rows && k < kc) {
                if (sum_mode) {
                    v = A1[(size_t)row * ldA1 + kg] +
                        ope * A2[(size_t)row * ldA2 + kg];
                } else {
                    if (kg < K1) v = A1[(size_t)row * ldA1 + kg];
                    else         v = ope * A2[(size_t)row * ldA2 + (kg - K1)];
                }
            }
            As[r * 68 + k] = v;
        }
        // --- stage W tile as K-pairs: Wp[k/2][n] = {W[k][n], W[k+1][n]} ---
        for (int idx = tid; idx < 32 * 64; idx += 128) {
            int k2 = idx >> 6, n = idx & 63;
            int k  = 2 * k2;
            float w0 = (k     < kc) ? W[(size_t)(k0 + k)     * Nfull + colbase + n] : 0.0f;
            float w1 = (k + 1 < kc) ? W[(size_t)(k0 + k + 1) * Nfull + colbase + n] : 0.0f;
            Wp[k2 * 80 + n] = make_float2(w0, w1);
        }
        __syncthreads();

        // --- WMMA over the chunk: K step 4, 4 column tiles per wave ---
        const int arow = (wave * 16 + lane16) * 68;
        for (int kk = 0; kk < kcp; kk += 4) {
            v2f a;
            a.x = As[arow + kk + 2 * hi];
            a.y = As[arow + kk + 2 * hi + 1];
            const int wrow = ((kk >> 1) + hi) * 80;
#pragma unroll
            for (int c = 0; c < 4; ++c) {
                float2 w2 = Wp[wrow + c * 16 + lane16];
                v2f bf;
                bf.x = w2.x;
                bf.y = w2.y;
                acc[c] = __builtin_amdgcn_wmma_f32_16x16x4_f32(
                    false, a, false, bf, (short)0, acc[c], false, false);
            }
        }
        __syncthreads();
    }

    // --- epilogue: bias + PReLU, optional accumulate ---
#pragma unroll
    for (int c = 0; c < 4; ++c) {
#pragma unroll
        for (int v = 0; v < 8; ++v) {
            int m   = wave * 16 + hi * 8 + v;
            int row = rowbase + m;
            if (row < rows) {
                int col = colbase + c * 16 + lane16;
                float val = acc[c][v] + bias[col];
                val = (val >= 0.0f) ? val : slope * val;
                size_t o = (size_t)row * ldo + col;
                if (accumulate) out[o] += val;
                else            out[o] = val;
            }
        }
    }
}

// ---------------------------------------------------------------------------
// MLP head: [rows,128] @ [128,1] + b  (N=1: plain per-row dot product)
// ---------------------------------------------------------------------------
__global__ void head_kernel(const float* __restrict__ h,
                            const float* __restrict__ W,
                            const float* __restrict__ b,
                            float* __restrict__ out, int rows) {
    int r = blockIdx.x * blockDim.x + threadIdx.x;
    if (r >= rows) return;
    float s = b[0];
    const float* hp = h + (size_t)r * 128;
#pragma unroll
    for (int k = 0; k < 128; k += 4) {
        float4 v = *(const float4*)(hp + k);
        s += v.x * W[k] + v.y * W[k + 1] + v.z * W[k + 2] + v.w * W[k + 3];
    }
    out[r] = s;
}

// ---------------------------------------------------------------------------
// Input index map (top level in setup_inputs() dict order; nested params
// flattened JAX-style: dict keys sorted, lists in order):
//  0 x_path  1 x_link  2 x_node
//  3 ei_uses 4 ei_includes 5 ei_connects 6 ei_has  7 path_batch
//  params["layers"][l] -> relations sorted: connects,has,includes,uses
//    each relation sorted: W,a,b,eps  ('W' < 'a' in ASCII)
//    base(l) = 8 + 16*l; connects=+0, has=+4, includes=+8, uses=+12
//  params["mlp"][j]: W,b at 56 + 2*j
// ---------------------------------------------------------------------------
extern "C" void kernel_launch(void* const* d_in, const int* in_sizes, int n_in,
                              void* d_out, int out_size, void* d_ws, size_t ws_size,
                              hipStream_t stream) {
    (void)in_sizes; (void)n_in; (void)out_size; (void)ws_size;

    const float* x_path = (const float*)d_in[0];
    const float* x_link = (const float*)d_in[1];
    const float* x_node = (const float*)d_in[2];
    const int* ei_uses     = (const int*)d_in[3];
    const int* ei_includes = (const int*)d_in[4];
    const int* ei_connects = (const int*)d_in[5];
    const int* ei_has      = (const int*)d_in[6];

    auto F = [&](int i) { return (const float*)d_in[i]; };

    // ---------------- workspace layout (floats) ----------------
    const size_t PD = (size_t)P_CNT * D_EMB;   // 12.8M
    const size_t LD = (size_t)L_CNT * D_EMB;   //  6.4M
    const size_t ND = (size_t)N_CNT * D_EMB;   //  1.28M
    float* ws   = (float*)d_ws;
    float* xp_a = ws;            // layer1 output
    float* xp_b = ws + PD;       // layer0 & layer2 (final) path features
    float* R    = ws + 2 * PD;   // reusable region
    float* xl_a   = R;
    float* xl_b   = xl_a + LD;
    float* xn_a   = xl_b + LD;
    float* xn_b   = xn_a + ND;
    float* agg_l1 = xn_b + ND;          // L x 64
    float* agg_l2 = agg_l1 + LD;        // L x 64
    float* agg_p  = agg_l2 + LD;        // P x 64
    float* agg_n  = agg_p + PD;         // N x 64
    float* h1 = R;                      // P x 256 (aliases dead GNN buffers)
    float* h2 = h1 + (size_t)P_CNT * 256;  // P x 128

    auto scatter = [&](const float* src, int f, const int* ei, int e,
                       float* agg, int ndst) {
        hipMemsetAsync(agg, 0, (size_t)ndst * f * sizeof(float), stream);
        if (f == 64) {
            long long total = (long long)e * 16;
            int blocks = (int)((total + 255) / 256);
            scatter_add64_kernel<<<blocks, 256, 0, stream>>>(src, ei, e, agg);
        } else {
            long long total = (long long)e * f;
            int blocks = (int)((total + 255) / 256);
            scatter_add_kernel<<<blocks, 256, 0, stream>>>(src, f, ei, e, agg);
        }
    };

    auto gemm = [&](const float* A1, int ldA1, int K1,
                    const float* A2, int ldA2, int K2,
                    const float* eps_p, int sum_mode,
                    const float* Wp, int Nfull, const float* bias,
                    const float* slope_p, float slope_c,
                    float* out, int ldo, int rows, int accum) {
        dim3 grid((rows + 63) / 64, Nfull / 64);
        gemm_wmma_kernel<<<grid, 128, 0, stream>>>(
            A1, ldA1, K1, A2, ldA2, K2, eps_p, sum_mode,
            Wp, Nfull, bias, slope_p, slope_c, out, ldo, rows, accum);
    };

    // ---------------- layer 0 (concat mode, raw input features) ----------------
    // relation param indices: base=8; connects 8..11, has 12..15, includes 16..19, uses 20..23
    scatter(x_path, 9,  ei_uses,     EPL, agg_l1, L_CNT);   // path -> link
    scatter(x_node, 4,  ei_has,      ELN, agg_l2, L_CNT);   // node -> link
    scatter(x_link, 10, ei_includes, EPL, agg_p,  P_CNT);   // link -> path
    scatter(x_link, 10, ei_connects, ELN, agg_n,  N_CNT);   // link -> node

    // new_l = prelu([agg_u,(1+e)xl] Wu + bu) + prelu([agg_h,(1+e)xl] Wh + bh)
    gemm(agg_l1, 9,  9,  x_link, 10, 10, F(23), 0, F(20), 64, F(22), F(21), 0.f, xl_b, 64, L_CNT, 0);
    gemm(agg_l2, 4,  4,  x_link, 10, 10, F(15), 0, F(12), 64, F(14), F(13), 0.f, xl_b, 64, L_CNT, 1);
    gemm(agg_p, 10, 10,  x_path,  9,  9, F(19), 0, F(16), 64, F(18), F(17), 0.f, xp_b, 64, P_CNT, 0);
    gemm(agg_n, 10, 10,  x_node,  4,  4, F(11), 0, F( 8), 64, F(10), F( 9), 0.f, xn_b, 64, N_CNT, 0);

    // ---------------- layers 1,2 (sum mode, 64->64) ----------------
    const float* xp_i = xp_b; const float* xl_i = xl_b; const float* xn_i = xn_b;
    float* xp_o = xp_a; float* xl_o = xl_a; float* xn_o = xn_a;
    for (int l = 1; l < 3; ++l) {
        int base = 8 + 16 * l;  // connects=+0, has=+4, includes=+8, uses=+12
        scatter(xp_i, 64, ei_uses,     EPL, agg_l1, L_CNT);
        scatter(xn_i, 64, ei_has,      ELN, agg_l2, L_CNT);
        scatter(xl_i, 64, ei_includes, EPL, agg_p,  P_CNT);
        scatter(xl_i, 64, ei_connects, ELN, agg_n,  N_CNT);

        gemm(agg_l1, 64, 64, xl_i, 64, 64, F(base+15), 1, F(base+12), 64, F(base+14), F(base+13), 0.f, xl_o, 64, L_CNT, 0);
        gemm(agg_l2, 64, 64, xl_i, 64, 64, F(base+7),  1, F(base+4),  64, F(base+6),  F(base+5),  0.f, xl_o, 64, L_CNT, 1);
        gemm(agg_p,  64, 64, xp_i, 64, 64, F(base+11), 1, F(base+8),  64, F(base+10), F(base+9),  0.f, xp_o, 64, P_CNT, 0);
        gemm(agg_n,  64, 64, xn_i, 64, 64, F(base+3),  1, F(base+0),  64, F(base+2),  F(base+1),  0.f, xn_o, 64, N_CNT, 0);

        // swap ping-pong: layer1 wrote _a, layer2 reads _a writes _b
        const float* tp;
        tp = xp_i; xp_i = xp_o; xp_o = (float*)tp;
        tp = xl_i; xl_i = xl_o; xl_o = (float*)tp;
        tp = xn_i; xn_i = xn_o; xn_o = (float*)tp;
    }
    // final path features in xp_i (== xp_b after 2 swaps)

    // ---------------- MLP readout ----------------
    // h1 = relu([xp, x_path] @ W0 + b0)   (73 -> 256), concat mode with ope=1
    gemm(xp_i, 64, 64, x_path, 9, 9, nullptr, 0, F(56), 256, F(57), nullptr, 0.0f, h1, 256, P_CNT, 0);
    // h2 = relu(h1 @ W1 + b1)             (256 -> 128)
    gemm(h1, 256, 256, nullptr, 0, 0, nullptr, 0, F(58), 128, F(59), nullptr, 0.0f, h2, 128, P_CNT, 0);
    // out = h2 @ W2 + b2                  (128 -> 1)
    head_kernel<<<(P_CNT + 255) / 256, 256, 0, stream>>>(h2, F(60), F(61), (float*)d_out, P_CNT);
}